// SAGEResidualReranker_48885317763315
// MI455X (gfx1250) — compile-verified
//
#include <hip/hip_runtime.h>
#include <hip/hip_bf16.h>

// ---------------------------------------------------------------------------
// SAGEResidualReranker for MI455X (gfx1250, wave32)
//   h1 = relu( (scatter_mean(x)) @ Wl1^T + bl1 + x @ Wr1^T )
//   h2 = relu( (scatter_mean(h1)) @ Wl2^T + bl2 + h1 @ Wr2^T ) + x
//   out = sigmoid(alpha)*rer + (1-sigmoid(alpha))*(h2 @ Ws^T + bs)
// GEMMs use V_WMMA_F32_16X16X4_F32 (exact f32 matrix math on CDNA5).
// ---------------------------------------------------------------------------

typedef __attribute__((ext_vector_type(2))) float v2f;
typedef __attribute__((ext_vector_type(8))) float v8f;

#define D_FEAT 128

// ---------------- zero fill -------------------------------------------------
__global__ void zero_f32_kernel(float* __restrict__ p, long long n) {
    long long i = (long long)blockIdx.x * blockDim.x + threadIdx.x;
    long long stride = (long long)gridDim.x * blockDim.x;
    for (; i < n; i += stride) p[i] = 0.0f;
}

// ---------------- degree count ---------------------------------------------
__global__ void count_kernel(const int* __restrict__ dst, float* __restrict__ cnt, int E) {
    int e = blockIdx.x * blockDim.x + threadIdx.x;
    if (e < E) atomicAdd(&cnt[dst[e]], 1.0f);
}

// ---------------- edge scatter: agg[dst] += feat[src] ----------------------
// One wave (32 lanes) per edge; each lane moves a float4 (4 features).
__global__ void scatter_kernel(const float* __restrict__ feat,
                               const int* __restrict__ src,
                               const int* __restrict__ dst,
                               float* __restrict__ agg, int E) {
    long long tid = (long long)blockIdx.x * blockDim.x + threadIdx.x;
    int e = (int)(tid >> 5);
    if (e >= E) return;
    int lane = (int)(tid & 31);
    int s = src[e];
    int d = dst[e];
    const float4 v = ((const float4*)(feat + (long long)s * D_FEAT))[lane];
    float* bp = agg + (long long)d * D_FEAT + lane * 4;
    atomicAdd(bp + 0, v.x);
    atomicAdd(bp + 1, v.y);
    atomicAdd(bp + 2, v.z);
    atomicAdd(bp + 3, v.w);
}

// ---------------- mean normalize (true division, matches reference) --------
__global__ void normalize_kernel(float* __restrict__ agg, const float* __restrict__ cnt,
                                 long long total) {
    long long i = (long long)blockIdx.x * blockDim.x + threadIdx.x;
    if (i >= total) return;
    float c = fmaxf(cnt[i >> 7], 1.0f);   // 128 feats per row
    agg[i] = agg[i] / c;
}

// ---------------- fused dual-GEMM SAGE layer via WMMA f32 ------------------
// Out[r,c] = relu( sum_k Anorm[r,k]*Wl[c,k] + sum_k H[r,k]*Wr[c,k] + bias[c] )
//            (+ resid[r,c] if resid != nullptr)
// Block = 256 threads = 8 waves; wave w owns the 16x16 tile at cols [16w,16w+16).
// blockIdx.x selects the 16-row stripe.
__global__ void sage_gemm_wmma_kernel(const float* __restrict__ Anorm,
                                      const float* __restrict__ H,
                                      const float* __restrict__ Wl,
                                      const float* __restrict__ Wr,
                                      const float* __restrict__ bias,
                                      const float* __restrict__ resid,
                                      float* __restrict__ Out, int Nn) {
    const int wave = threadIdx.x >> 5;
    const int lane = threadIdx.x & 31;
    const int half = lane >> 4;       // 0: lanes 0-15, 1: lanes 16-31
    const int mn   = lane & 15;       // M index for A-frag, N index for B-frag
    const int r0   = blockIdx.x * 16;
    const int c0   = wave * 16;

    // clamp row for loads (uniform arithmetic, EXEC stays all-ones for WMMA)
    const int arow = (r0 + mn < Nn) ? (r0 + mn) : (Nn - 1);

    v8f acc = {};

    // ---- GEMM 1: Anorm @ Wl^T  (B[k][c] = Wl[c][k], Wl row-major [128][128])
    #pragma unroll 4
    for (int k0 = 0; k0 < D_FEAT; k0 += 4) {
        const int kk = k0 + 2 * half;
        // A fragment: 16x4 f32.  VGPR0/lo=K0, VGPR1/lo=K1, VGPR0/hi=K2, VGPR1/hi=K3
        const float* ap = Anorm + (long long)arow * D_FEAT + kk;
        v2f a; a.x = ap[0]; a.y = ap[1];
        // B fragment: 4x16 f32, mirrored layout (lane = N column)
        const float* bp = Wl + (long long)(c0 + mn) * D_FEAT + kk;
        v2f b; b.x = bp[0]; b.y = bp[1];
        acc = __builtin_amdgcn_wmma_f32_16x16x4_f32(false, a, false, b,
                                                    (short)0, acc, false, false);
    }

    // ---- GEMM 2: H @ Wr^T
    #pragma unroll 4
    for (int k0 = 0; k0 < D_FEAT; k0 += 4) {
        const int kk = k0 + 2 * half;
        const float* ap = H + (long long)arow * D_FEAT + kk;
        v2f a; a.x = ap[0]; a.y = ap[1];
        const float* bp = Wr + (long long)(c0 + mn) * D_FEAT + kk;
        v2f b; b.x = bp[0]; b.y = bp[1];
        acc = __builtin_amdgcn_wmma_f32_16x16x4_f32(false, a, false, b,
                                                    (short)0, acc, false, false);
    }

    // ---- epilogue: bias, relu, optional residual, store
    const float bc = bias[c0 + mn];
    #pragma unroll
    for (int v = 0; v < 8; ++v) {
        const int row = r0 + v + 8 * half;   // C/D layout: VGPR v -> M = v (+8 for hi lanes)
        if (row < Nn) {
            float val = fmaxf(acc[v] + bc, 0.0f);
            const long long idx = (long long)row * D_FEAT + c0 + mn;
            if (resid) val += resid[idx];
            Out[idx] = val;
        }
    }
}

// ---------------- score head + blend ---------------------------------------
__global__ void score_kernel(const float* __restrict__ h,
                             const float* __restrict__ rer,
                             const float* __restrict__ Ws,
                             const float* __restrict__ bs,
                             const float* __restrict__ alpha_p,
                             float* __restrict__ out, int Nn) {
    int n = blockIdx.x * blockDim.x + threadIdx.x;
    if (n >= Nn) return;
    const float4* hp = (const float4*)(h + (long long)n * D_FEAT);
    const float4* wp = (const float4*)Ws;
    float dot = 0.0f;
    #pragma unroll
    for (int i = 0; i < D_FEAT / 4; ++i) {
        float4 a = hp[i], b = wp[i];
        dot += a.x * b.x + a.y * b.y + a.z * b.z + a.w * b.w;
    }
    float a = 1.0f / (1.0f + expf(-alpha_p[0]));
    out[n] = a * rer[n] + (1.0f - a) * (dot + bs[0]);
}

// ---------------------------------------------------------------------------
extern "C" void kernel_launch(void* const* d_in, const int* in_sizes, int n_in,
                              void* d_out, int out_size, void* d_ws, size_t ws_size,
                              hipStream_t stream) {
    const float* x     = (const float*)d_in[0];
    const int*   eidx  = (const int*)d_in[1];
    const float* rer   = (const float*)d_in[2];
    const float* Wl1   = (const float*)d_in[3];
    const float* bl1   = (const float*)d_in[4];
    const float* Wr1   = (const float*)d_in[5];
    const float* Wl2   = (const float*)d_in[6];
    const float* bl2   = (const float*)d_in[7];
    const float* Wr2   = (const float*)d_in[8];
    const float* Ws    = (const float*)d_in[9];
    const float* bs    = (const float*)d_in[10];
    const float* alpha = (const float*)d_in[11];

    const int Nn = in_sizes[2];          // 50000 (reranker_scores length)
    const int E  = in_sizes[1] / 2;      // 800000
    const int* src = eidx;
    const int* dst = eidx + E;

    float* out = (float*)d_out;

    // workspace layout: agg | cnt | h1 | h2
    float* agg = (float*)d_ws;
    float* cnt = agg + (long long)Nn * D_FEAT;
    float* h1  = cnt + Nn;
    float* h2  = h1 + (long long)Nn * D_FEAT;

    const long long featElems = (long long)Nn * D_FEAT;

    dim3 blk(256);
    const int zeroGrid  = 2048;
    const int cntGrid   = (E + 255) / 256;
    const int scatGrid  = (int)(((long long)E * 32 + 255) / 256);
    const int normGrid  = (int)((featElems + 255) / 256);
    const int gemmGrid  = (Nn + 15) / 16;   // 3125 stripes, 8 waves each
    const int scoreGrid = (Nn + 255) / 256;

    // ---- layer 1 ----
    zero_f32_kernel<<<zeroGrid, blk, 0, stream>>>(agg, featElems + Nn); // agg + cnt
    count_kernel<<<cntGrid, blk, 0, stream>>>(dst, cnt, E);
    scatter_kernel<<<scatGrid, blk, 0, stream>>>(x, src, dst, agg, E);
    normalize_kernel<<<normGrid, blk, 0, stream>>>(agg, cnt, featElems);
    sage_gemm_wmma_kernel<<<gemmGrid, blk, 0, stream>>>(agg, x, Wl1, Wr1, bl1,
                                                        nullptr, h1, Nn);

    // ---- layer 2 ----
    zero_f32_kernel<<<zeroGrid, blk, 0, stream>>>(agg, featElems);
    scatter_kernel<<<scatGrid, blk, 0, stream>>>(h1, src, dst, agg, E);
    normalize_kernel<<<normGrid, blk, 0, stream>>>(agg, cnt, featElems);
    sage_gemm_wmma_kernel<<<gemmGrid, blk, 0, stream>>>(agg, h1, Wl2, Wr2, bl2,
                                                        x /*residual*/, h2, Nn);

    // ---- score head + reranker blend ----
    score_kernel<<<scoreGrid, blk, 0, stream>>>(h2, rer, Ws, bs, alpha, out, Nn);
}